// GAT_49864570307071
// MI455X (gfx1250) — compile-verified
//
#include <hip/hip_runtime.h>
#include <hip/hip_bf16.h>

typedef float v2f __attribute__((ext_vector_type(2)));
typedef float v8f __attribute__((ext_vector_type(8)));

// ---------------------------------------------------------------------------
// Monotonic float<->uint encoding so segment-max can use GLOBAL_ATOMIC_MAX_U32
// ---------------------------------------------------------------------------
__device__ __forceinline__ unsigned fenc(float f) {
    unsigned u = __float_as_uint(f);
    return (u & 0x80000000u) ? ~u : (u | 0x80000000u);
}
__device__ __forceinline__ float fdec(unsigned u) {
    return __uint_as_float((u & 0x80000000u) ? (u ^ 0x80000000u) : ~u);
}
__device__ __forceinline__ float lrelu(float v) { return v > 0.f ? v : 0.2f * v; }

// ---------------------------------------------------------------------------
// Dense projection H = X @ W  via V_WMMA_F32_16X16X4_F32 (full f32 precision).
// X: [N,128] row-major, W: [128,128] row-major, H: [N,128].
// One wave per 16x16 output tile; block = 256 threads = 8 waves = 8 col tiles.
// N % 16 == 0 and dims are exact -> no guards, EXEC stays all-ones for WMMA.
// ---------------------------------------------------------------------------
__global__ void __launch_bounds__(256)
gat_gemm_wmma_f32(const float* __restrict__ X, const float* __restrict__ W,
                  float* __restrict__ Hout) {
    const int lane = threadIdx.x & 31;
    const int wv   = threadIdx.x >> 5;          // 0..7 : column tile
    const int row0 = blockIdx.x * 16;
    const int col0 = wv * 16;
    const int mn   = lane & 15;                 // row (A) / col (B,C/D) in tile
    const int kb   = (lane >> 4) << 1;          // K base: 0 or 2

    // Preload whole 128x16 B panel into registers (64 VGPRs).
    v2f Breg[32];
#pragma unroll
    for (int kk = 0; kk < 32; ++kk) {
        const int k = kk * 4 + kb;
        Breg[kk].x = W[(size_t)k * 128 + col0 + mn];
        Breg[kk].y = W[(size_t)(k + 1) * 128 + col0 + mn];
    }

    v8f c = {};
    const float* arow = X + (size_t)(row0 + mn) * 128 + kb;
#pragma unroll
    for (int kk = 0; kk < 32; ++kk) {
        v2f a = *(const v2f*)(arow + kk * 4);   // contiguous K,K+1 -> b64 load
        c = __builtin_amdgcn_wmma_f32_16x16x4_f32(
                false, a, false, Breg[kk], (short)0, c, false, false);
    }

    float* orow = Hout + (size_t)(row0 + ((lane >> 4) << 3)) * 128 + col0 + mn;
#pragma unroll
    for (int v = 0; v < 8; ++v) orow[(size_t)v * 128] = c[v];
}

// ---------------------------------------------------------------------------
// Per-(node,head) attention logits: alS[n,h] = h[n,h,:]·a_src[h,:], same for dst.
// ---------------------------------------------------------------------------
__global__ void gat_node_logits(const float* __restrict__ Hfeat,
                                const float* __restrict__ a_src,
                                const float* __restrict__ a_dst,
                                float* __restrict__ alS, float* __restrict__ alD,
                                int Ncnt) {
    const int id = blockIdx.x * blockDim.x + threadIdx.x;
    if (id >= Ncnt * 4) return;
    const int n = id >> 2, h = id & 3;
    const float* hr = Hfeat + (size_t)n * 128 + h * 32;
    const float* as = a_src + h * 32;
    const float* ad = a_dst + h * 32;
    float ss = 0.f, sd = 0.f;
#pragma unroll
    for (int i = 0; i < 32; i += 4) {
        const float4 hv = *(const float4*)(hr + i);
        const float4 sv = *(const float4*)(as + i);
        const float4 dv = *(const float4*)(ad + i);
        ss += hv.x * sv.x + hv.y * sv.y + hv.z * sv.z + hv.w * sv.w;
        sd += hv.x * dv.x + hv.y * dv.y + hv.z * dv.z + hv.w * dv.w;
    }
    alS[id] = ss;
    alD[id] = sd;
}

// ---------------------------------------------------------------------------
// Init per layer: agg = bias (broadcast), segment-max = -inf (enc 0), denom = 0.
// ---------------------------------------------------------------------------
__global__ void gat_init_layer(float* __restrict__ agg, const float* __restrict__ bias,
                               unsigned* __restrict__ menc, float* __restrict__ denom,
                               int Ncnt) {
    const int i = blockIdx.x * blockDim.x + threadIdx.x;
    if (i < Ncnt * 128) agg[i] = bias[i & 127];
    if (i < Ncnt * 4) { menc[i] = 0u; denom[i] = 0.f; }
}

// ---------------------------------------------------------------------------
// Edge pass A: e = leakyrelu(alS[src] + alD[dst]); store; atomicMax segment max.
// Self-loops are edges [E, E+N).
// ---------------------------------------------------------------------------
__global__ void gat_edge_logits_max(const int* __restrict__ ei, int E, int Etot,
                                    const float* __restrict__ alS,
                                    const float* __restrict__ alD,
                                    float* __restrict__ ebuf,
                                    unsigned* __restrict__ menc) {
    const int e = blockIdx.x * blockDim.x + threadIdx.x;
    if (e >= Etot) return;
    int s, d;
    if (e < E) { s = ei[e]; d = ei[E + e]; } else { s = e - E; d = s; }
    const float4 as = *(const float4*)(alS + (size_t)s * 4);
    const float4 ad = *(const float4*)(alD + (size_t)d * 4);
    float4 lv;
    lv.x = lrelu(as.x + ad.x);
    lv.y = lrelu(as.y + ad.y);
    lv.z = lrelu(as.z + ad.z);
    lv.w = lrelu(as.w + ad.w);
    *(float4*)(ebuf + (size_t)e * 4) = lv;
    unsigned* mp = menc + (size_t)d * 4;
    atomicMax(mp + 0, fenc(lv.x));
    atomicMax(mp + 1, fenc(lv.y));
    atomicMax(mp + 2, fenc(lv.z));
    atomicMax(mp + 3, fenc(lv.w));
}

// ---------------------------------------------------------------------------
// Edge pass B: ex = exp(e - m[dst]); store in place; atomicAdd denominator.
// ---------------------------------------------------------------------------
__global__ void gat_edge_exp_sum(const int* __restrict__ ei, int E, int Etot,
                                 float* __restrict__ ebuf,
                                 const unsigned* __restrict__ menc,
                                 float* __restrict__ denom) {
    const int e = blockIdx.x * blockDim.x + threadIdx.x;
    if (e >= Etot) return;
    const int d = (e < E) ? ei[E + e] : (e - E);
    float4 lv = *(const float4*)(ebuf + (size_t)e * 4);
    const uint4 mu = *(const uint4*)(menc + (size_t)d * 4);
    float4 ex;
    ex.x = __expf(lv.x - fdec(mu.x));
    ex.y = __expf(lv.y - fdec(mu.y));
    ex.z = __expf(lv.z - fdec(mu.z));
    ex.w = __expf(lv.w - fdec(mu.w));
    *(float4*)(ebuf + (size_t)e * 4) = ex;
    float* dp = denom + (size_t)d * 4;
    atomicAdd(dp + 0, ex.x);
    atomicAdd(dp + 1, ex.y);
    atomicAdd(dp + 2, ex.z);
    atomicAdd(dp + 3, ex.w);
}

// ---------------------------------------------------------------------------
// Edge pass C: one wave per edge; lane handles 4 channels (head = lane>>3).
// out[dst] += (ex/denom[dst]) * h[src]  via f32 global atomics.
// ---------------------------------------------------------------------------
__global__ void gat_edge_aggregate(const int* __restrict__ ei, int E, int Etot,
                                   const float* __restrict__ ebuf,
                                   const float* __restrict__ denom,
                                   const float* __restrict__ Hfeat,
                                   float* __restrict__ out) {
    const int wave = (int)((blockIdx.x * (size_t)blockDim.x + threadIdx.x) >> 5);
    const int lane = threadIdx.x & 31;
    if (wave >= Etot) return;
    int s, d;
    if (wave < E) { s = ei[wave]; d = ei[E + wave]; } else { s = wave - E; d = s; }
    const int h = lane >> 3;
    const float alpha = ebuf[(size_t)wave * 4 + h] / denom[(size_t)d * 4 + h];
    const float4 hs = *(const float4*)(Hfeat + (size_t)s * 128 + lane * 4);
    float* o = out + (size_t)d * 128 + lane * 4;
    atomicAdd(o + 0, alpha * hs.x);
    atomicAdd(o + 1, alpha * hs.y);
    atomicAdd(o + 2, alpha * hs.z);
    atomicAdd(o + 3, alpha * hs.w);
}

// ---------------------------------------------------------------------------
// ELU (alpha=1) in place between the two layers.
// ---------------------------------------------------------------------------
__global__ void gat_elu_inplace(float* __restrict__ p, int total) {
    const int i = blockIdx.x * blockDim.x + threadIdx.x;
    if (i >= total) return;
    const float v = p[i];
    p[i] = v > 0.f ? v : expm1f(v);
}

// ---------------------------------------------------------------------------
// Host-side orchestration (graph-capture safe: only launches on `stream`).
// ---------------------------------------------------------------------------
extern "C" void kernel_launch(void* const* d_in, const int* in_sizes, int n_in,
                              void* d_out, int out_size, void* d_ws, size_t ws_size,
                              hipStream_t stream) {
    (void)n_in; (void)out_size; (void)ws_size;
    const float* x      = (const float*)d_in[0];
    const int*   ei     = (const int*)d_in[1];
    const float* W1     = (const float*)d_in[2];
    const float* a_src1 = (const float*)d_in[3];
    const float* a_dst1 = (const float*)d_in[4];
    const float* b1     = (const float*)d_in[5];
    const float* W2     = (const float*)d_in[6];
    const float* a_src2 = (const float*)d_in[7];
    const float* a_dst2 = (const float*)d_in[8];
    const float* b2     = (const float*)d_in[9];
    float* outp = (float*)d_out;

    const int N    = in_sizes[0] / 128;   // 50000
    const int E    = in_sizes[1] / 2;     // 800000
    const int Etot = E + N;               // + self loops

    // Workspace layout (all 16B aligned)
    char* ws = (char*)d_ws;
    float*    Hbuf  = (float*)ws;            ws += (size_t)N * 128 * 4;
    float*    agg1  = (float*)ws;            ws += (size_t)N * 128 * 4;
    float*    alS   = (float*)ws;            ws += (size_t)N * 4 * 4;
    float*    alD   = (float*)ws;            ws += (size_t)N * 4 * 4;
    unsigned* menc  = (unsigned*)ws;         ws += (size_t)N * 4 * 4;
    float*    denom = (float*)ws;            ws += (size_t)N * 4 * 4;
    float*    ebuf  = (float*)ws;            ws += (size_t)Etot * 4 * 4;

    const int BLK = 256;
    const dim3 gGemm(N / 16);                               // 3125 tiles, exact
    const dim3 gNode((N * 4 + BLK - 1) / BLK);
    const dim3 gInit((N * 128 + BLK - 1) / BLK);
    const dim3 gEdge((Etot + BLK - 1) / BLK);
    const dim3 gAggr(((size_t)Etot * 32 + BLK - 1) / BLK);
    const dim3 gElu((N * 128 + BLK - 1) / BLK);

    // ---------------- Layer 1 ----------------
    gat_gemm_wmma_f32<<<gGemm, BLK, 0, stream>>>(x, W1, Hbuf);
    gat_node_logits<<<gNode, BLK, 0, stream>>>(Hbuf, a_src1, a_dst1, alS, alD, N);
    gat_init_layer<<<gInit, BLK, 0, stream>>>(agg1, b1, menc, denom, N);
    gat_edge_logits_max<<<gEdge, BLK, 0, stream>>>(ei, E, Etot, alS, alD, ebuf, menc);
    gat_edge_exp_sum<<<gEdge, BLK, 0, stream>>>(ei, E, Etot, ebuf, menc, denom);
    gat_edge_aggregate<<<gAggr, BLK, 0, stream>>>(ei, E, Etot, ebuf, denom, Hbuf, agg1);
    gat_elu_inplace<<<gElu, BLK, 0, stream>>>(agg1, N * 128);

    // ---------------- Layer 2 ----------------
    gat_gemm_wmma_f32<<<gGemm, BLK, 0, stream>>>(agg1, W2, Hbuf);
    gat_node_logits<<<gNode, BLK, 0, stream>>>(Hbuf, a_src2, a_dst2, alS, alD, N);
    gat_init_layer<<<gInit, BLK, 0, stream>>>(outp, b2, menc, denom, N);
    gat_edge_logits_max<<<gEdge, BLK, 0, stream>>>(ei, E, Etot, alS, alD, ebuf, menc);
    gat_edge_exp_sum<<<gEdge, BLK, 0, stream>>>(ei, E, Etot, ebuf, menc, denom);
    gat_edge_aggregate<<<gAggr, BLK, 0, stream>>>(ei, E, Etot, ebuf, denom, Hbuf, outp);
}